// YOLOLayer_30030411333645
// MI455X (gfx1250) — compile-verified
//
#include <hip/hip_runtime.h>
#include <hip/hip_bf16.h>

// ---------------------------------------------------------------------------
// YOLO head for MI455X (gfx1250).
// Memory-bound layout transform: 66.8MB in + 66.8MB out @ 23.3TB/s => ~5.7us.
// Strategy: per-block, async-copy a [128 pos x 85 feat] tile from global to
// LDS *transposed* (GLOBAL_LOAD_ASYNC_TO_LDS_B32: coalesced 128B global reads,
// per-lane LDS scatter with stride 85 words -> bank-conflict free), wait on
// ASYNCcnt, barrier, then emit fully-coalesced NT stores for boxes/conf/cls
// using 1-TRANS-op sigmoid/exp (v_exp_f32 + v_rcp_f32).
// ---------------------------------------------------------------------------

#define YB   16
#define YNA  3
#define YNC  80
#define YF   85            // 5 + NC features per anchor
#define YG   64
#define YGG  4096          // 64*64 positions
#define YTP  128           // positions staged per block
#define YNT  256           // threads per block (8 wave32)
#define YNPB (YNA * YGG)   // 12288 predictions per image

typedef float v4f __attribute__((ext_vector_type(4)));

#if defined(__AMDGCN__) && __has_builtin(__builtin_amdgcn_global_load_async_to_lds_b32)
#define YOLO_ASYNC_BUILTIN 1
#endif

__device__ __forceinline__ float fast_exp(float v) {
#if defined(__AMDGCN__)
    return __builtin_amdgcn_exp2f(v * 1.4426950408889634f);  // v_exp_f32
#else
    return expf(v);
#endif
}

__device__ __forceinline__ float fast_sigmoid(float v) {
#if defined(__AMDGCN__)
    return __builtin_amdgcn_rcpf(1.0f + fast_exp(-v));       // v_rcp_f32
#else
    return 1.0f / (1.0f + expf(-v));
#endif
}

__global__ __launch_bounds__(YNT) void yolo_head_kernel(
    const float* __restrict__ x,
    float* __restrict__ boxes,   // [B, 12288, 4]
    float* __restrict__ conf,    // [B, 12288, 1]
    float* __restrict__ cls)     // [B, 12288, 80]
{
    // Tile staged transposed: smem[pos][feat], row stride 85 (odd -> no bank
    // conflicts for any access pattern below). 43,520 bytes.
    __shared__ float smem[YTP * YF];

    const int tid  = threadIdx.x;
    const int blk  = blockIdx.x;
    const int tile = blk & 31;           // 32 tiles of 128 positions
    const int ba   = blk >> 5;
    const int bimg = ba / YNA;
    const int a    = ba - bimg * YNA;
    const int pos0 = tile * YTP;

    // x[bimg, a*85 + c, pos0 + p]
    const float* gsrc = x + ((bimg * 255 + a * YF) * YGG + pos0);

    // ---- Stage: global -> LDS (transposing scatter), tracked on ASYNCcnt ----
    for (int idx = tid; idx < YF * YTP; idx += YNT) {
        const int c = idx >> 7;          // channel 0..84 (128-float runs in memory)
        const int p = idx & 127;         // position within tile
#if defined(YOLO_ASYNC_BUILTIN)
        __builtin_amdgcn_global_load_async_to_lds_b32(
            (__attribute__((address_space(1))) int*)(gsrc + c * YGG + p),
            (__attribute__((address_space(3))) int*)&smem[p * YF + c],
            0, 0);
#elif defined(__AMDGCN__)
        {
            unsigned lds_off = (unsigned)(unsigned long)
                (__attribute__((address_space(3))) const void*)&smem[p * YF + c];
            unsigned voff = (unsigned)((c * YGG + p) * 4);
            asm volatile("global_load_async_to_lds_b32 %0, %1, %2"
                         :: "v"(lds_off), "v"(voff), "s"(gsrc)
                         : "memory");
        }
#else
        smem[p * YF + c] = gsrc[c * YGG + p];
#endif
    }
#if defined(__AMDGCN__)
#if __has_builtin(__builtin_amdgcn_s_wait_asynccnt)
    __builtin_amdgcn_s_wait_asynccnt(0);
#else
    asm volatile("s_wait_asynccnt 0" ::: "memory");
#endif
#endif
    __syncthreads();

    // Prediction index base for this tile: n = a*4096 + pos
    const size_t nbase = (size_t)bimg * YNPB + a * YGG + pos0;

    // anchors scaled by grid (64) and STRIDE (8); x8 is an exact fp scale.
    const float aw8 = (a == 0) ? 51.2f  : (a == 1) ? 153.6f : 307.2f;
    const float ah8 = (a == 0) ? 66.56f : (a == 1) ? 179.2f : 281.6f;

    // ---- boxes (waves 0-3: float4 coalesced) + conf (waves 4-7) ----
    const int p   = tid & 127;
    const int pos = pos0 + p;
    if (tid < YTP) {
        const float t0 = smem[p * YF + 0];
        const float t1 = smem[p * YF + 1];
        const float tw = smem[p * YF + 2];
        const float th = smem[p * YF + 3];
        v4f o;
        o.x = (fast_sigmoid(t0) + (float)(pos & 63)) * 8.0f;  // (sx + grid_x)*stride
        o.y = (fast_sigmoid(t1) + (float)(pos >> 6)) * 8.0f;  // (sy + grid_y)*stride
        o.z = fast_exp(tw) * aw8;                             // exp(w)*aw*stride
        o.w = fast_exp(th) * ah8;                             // exp(h)*ah*stride
        __builtin_nontemporal_store(o, (v4f*)boxes + (nbase + p));
    } else {
        __builtin_nontemporal_store(fast_sigmoid(smem[p * YF + 4]),
                                    conf + nbase + p);
    }

    // ---- cls: 128*80 contiguous floats, fully coalesced scalar stores ----
    float* clsrow = cls + nbase * YNC;
    for (unsigned i = tid; i < YTP * YNC; i += YNT) {
        const unsigned pp = i / YNC;
        const unsigned f  = i - pp * YNC;
        __builtin_nontemporal_store(fast_sigmoid(smem[pp * YF + 5 + f]),
                                    clsrow + i);
    }
}

extern "C" void kernel_launch(void* const* d_in, const int* in_sizes, int n_in,
                              void* d_out, int out_size, void* d_ws, size_t ws_size,
                              hipStream_t stream) {
    (void)in_sizes; (void)n_in; (void)out_size; (void)d_ws; (void)ws_size;
    const float* x = (const float*)d_in[0];
    float* out   = (float*)d_out;
    float* boxes = out;                                   // 16*12288*4 = 786432
    float* conf  = out + (size_t)YB * YNPB * 4;           // +786432
    float* cls   = out + (size_t)YB * YNPB * 5;           // +983040
    const int nblocks = YB * YNA * (YGG / YTP);           // 1536
    yolo_head_kernel<<<nblocks, YNT, 0, stream>>>(x, boxes, conf, cls);
}